// PointConvEncoder_36816459661393
// MI455X (gfx1250) — compile-verified
//
#include <hip/hip_runtime.h>
#include <hip/hip_bf16.h>

// MI455X / gfx1250, wave32. fp32 WMMA (V_WMMA_F32_16X16X4_F32) used for all
// neighborhood matmuls and the final GEMM — exact fp32 semantics.

typedef __attribute__((ext_vector_type(2))) float v2f;
typedef __attribute__((ext_vector_type(8))) float v8f;

#define LEAKF 0.2f
#define EPSF  1e-5f

__device__ __forceinline__ float lrelu(float x){ return x > 0.0f ? x : LEAKF * x; }

__device__ __forceinline__ v8f wmma4(v2f a, v2f b, v8f c){
  // D = A(16x4,f32) x B(4x16,f32) + C(16x16,f32)
  return __builtin_amdgcn_wmma_f32_16x16x4_f32(false, a, false, b, (short)0, c, false, false);
}

// channel accessor for grouped point features: [xyz_rel(3) | feats(nf) | zero pad]
__device__ __forceinline__ float gchan(int c, float gx, float gy, float gz,
                                       const float* __restrict__ fr, int nf){
  if (c == 0) return gx;
  if (c == 1) return gy;
  if (c == 2) return gz;
  if (c < 3 + nf) return fr[c - 3];
  return 0.0f;
}

// ---------------------------------------------------------------------------
// Kernel 1: transpose point_cloud -> xyz0 (B,1024,3) and pointwise MLP:
// f = lrelu(LN(xyz0@W0+b0)); f = lrelu(LN(f@W1+b1))  -> feats (B,1024,16)
// ---------------------------------------------------------------------------
__global__ __launch_bounds__(256) void k_pointwise(
    const float* __restrict__ pc,
    const float* __restrict__ W0, const float* __restrict__ b0,
    const float* __restrict__ g0, const float* __restrict__ be0,
    const float* __restrict__ W1, const float* __restrict__ b1,
    const float* __restrict__ g1, const float* __restrict__ be1,
    float* __restrict__ xyz0, float* __restrict__ feats)
{
  __shared__ float sW0[48], sb0[16], sg0[16], sbe0[16];
  __shared__ float sW1[256], sb1[16], sg1[16], sbe1[16];
  int t = threadIdx.x;
  if (t < 48) sW0[t] = W0[t];
  if (t < 16){ sb0[t]=b0[t]; sg0[t]=g0[t]; sbe0[t]=be0[t];
               sb1[t]=b1[t]; sg1[t]=g1[t]; sbe1[t]=be1[t]; }
  sW1[t] = W1[t];
  __syncthreads();
  int gid = blockIdx.x*256 + t;          // B*N = 262144 threads
  int b = gid >> 10, n = gid & 1023;
  float x = pc[(b*3+0)*1024+n];
  float y = pc[(b*3+1)*1024+n];
  float z = pc[(b*3+2)*1024+n];
  float* xo = xyz0 + (size_t)(b*1024+n)*3;
  xo[0]=x; xo[1]=y; xo[2]=z;

  float h[16]; float s1=0.f, s2=0.f;
  #pragma unroll
  for (int o=0;o<16;++o){
    float v = x*sW0[o] + y*sW0[16+o] + z*sW0[32+o] + sb0[o];
    h[o]=v; s1+=v; s2+=v*v;
  }
  float mu = s1*(1.f/16.f);
  float inv = rsqrtf(s2*(1.f/16.f) - mu*mu + EPSF);
  #pragma unroll
  for (int o=0;o<16;++o) h[o] = lrelu((h[o]-mu)*inv*sg0[o] + sbe0[o]);

  float h2[16]; s1=0.f; s2=0.f;
  #pragma unroll
  for (int o=0;o<16;++o){
    float v = sb1[o];
    #pragma unroll
    for (int c=0;c<16;++c) v += h[c]*sW1[c*16+o];
    h2[o]=v; s1+=v; s2+=v*v;
  }
  mu = s1*(1.f/16.f);
  inv = rsqrtf(s2*(1.f/16.f) - mu*mu + EPSF);
  float* fo = feats + (size_t)(b*1024+n)*16;
  #pragma unroll
  for (int o=0;o<16;++o) fo[o] = lrelu((h2[o]-mu)*inv*sg1[o] + sbe1[o]);
}

// ---------------------------------------------------------------------------
// Kernel 2: kNN(K=16) of queries xyz0[:, ::2] (512) against xyz0 (1024).
// Stable ties (strict <) matching lax.top_k index order.
// ---------------------------------------------------------------------------
__global__ __launch_bounds__(256) void k_knn1(const float* __restrict__ xyz0,
                                              int* __restrict__ idx1)
{
  __shared__ float sx[1024], sy[1024], sz[1024];
  int b = blockIdx.x, t = threadIdx.x;
  for (int j=t;j<1024;j+=256){
    const float* p = xyz0 + (size_t)(b*1024+j)*3;
    sx[j]=p[0]; sy[j]=p[1]; sz[j]=p[2];
  }
  __syncthreads();
  #pragma unroll
  for (int q=0;q<2;++q){
    int m = t + q*256;                       // 0..511
    float qx=sx[2*m], qy=sy[2*m], qz=sz[2*m];
    float bd[16]; int bi[16];
    #pragma unroll
    for (int k=0;k<16;++k){ bd[k]=3.402823466e38f; bi[k]=0; }
    for (int j=0;j<1024;++j){
      float dx=sx[j]-qx, dy=sy[j]-qy, dz=sz[j]-qz;
      float d2 = dx*dx + dy*dy + dz*dz;
      if (d2 < bd[15]){
        bd[15]=d2; bi[15]=j;
        #pragma unroll
        for (int u=14;u>=0;--u){
          if (bd[u+1] < bd[u]){
            float td=bd[u]; bd[u]=bd[u+1]; bd[u+1]=td;
            int ti=bi[u]; bi[u]=bi[u+1]; bi[u+1]=ti;
          }
        }
      }
    }
    int* op = idx1 + (size_t)(b*512+m)*16;
    #pragma unroll
    for (int k=0;k<16;++k) op[k]=bi[k];
  }
}

// ---------------------------------------------------------------------------
// Kernel 3 (WMMA): pointconv1 + fused squeeze.
// One wave per (b,m): A = g (16 neighbors x 19ch, padded to 20),
// B = pc0_W (19x32) -> 2 n-tiles, 5 k-steps of f32 WMMA.
// lrelu, max over neighbors (M), then fs = lrelu(f@Ws+bs) (32->8).
// ---------------------------------------------------------------------------
__global__ __launch_bounds__(128) void k_pc1(
    const float* __restrict__ xyz0, const float* __restrict__ feats,
    const int* __restrict__ idx1,
    const float* __restrict__ W, const float* __restrict__ bias,
    const float* __restrict__ Ws, const float* __restrict__ bs,
    float* __restrict__ fs)
{
  __shared__ float sW[20*32];    // K padded to 20, zero rows
  __shared__ float sWs[32*8];
  __shared__ float sbias[32];
  __shared__ float sbs[8];
  __shared__ float sf[4][32];
  int t = threadIdx.x;
  for (int i=t;i<640;i+=128){ int c=i>>5, o=i&31; sW[i] = (c<19) ? W[c*32+o] : 0.0f; }
  for (int i=t;i<256;i+=128) sWs[i]=Ws[i];
  if (t<32) sbias[t]=bias[t];
  if (t<8)  sbs[t]=bs[t];
  __syncthreads();

  int wv = t>>5, lane = t&31;
  int gw = blockIdx.x*4 + wv;              // 0..B*512-1
  int b = gw>>9, m = gw&511;
  int row = lane & 15;                     // neighbor index (A-row M)
  int kh  = (lane>>4)<<1;                  // K sub-offset: 0 or 2
  int n   = lane & 15;                     // B/C column within tile

  const float* ctr = xyz0 + (size_t)(b*1024 + 2*m)*3;
  float cx=ctr[0], cy=ctr[1], cz=ctr[2];
  int nb = idx1[(size_t)(b*512+m)*16 + row];
  const float* pn = xyz0  + (size_t)(b*1024+nb)*3;
  const float* fr = feats + (size_t)(b*1024+nb)*16;
  float gx=pn[0]-cx, gy=pn[1]-cy, gz=pn[2]-cz;

  v8f a0, a1;
  float bv0 = sbias[n], bv1 = sbias[16+n];
  #pragma unroll
  for (int i=0;i<8;++i){ a0[i]=bv0; a1[i]=bv1; }

  #pragma unroll
  for (int s=0;s<5;++s){
    int c0 = 4*s + kh;
    v2f av; av.x = gchan(c0,   gx,gy,gz, fr, 16);
            av.y = gchan(c0+1, gx,gy,gz, fr, 16);
    v2f b0v, b1v;
    b0v.x = sW[c0*32 + n];        b0v.y = sW[(c0+1)*32 + n];
    b1v.x = sW[c0*32 + 16 + n];   b1v.y = sW[(c0+1)*32 + 16 + n];
    a0 = wmma4(av, b0v, a0);
    a1 = wmma4(av, b1v, a1);
  }

  float m0=-3.402823466e38f, m1=-3.402823466e38f;
  #pragma unroll
  for (int i=0;i<8;++i){ m0 = fmaxf(m0, lrelu(a0[i])); m1 = fmaxf(m1, lrelu(a1[i])); }
  m0 = fmaxf(m0, __shfl_xor(m0, 16, 32));   // combine M halves (lane L <-> L^16)
  m1 = fmaxf(m1, __shfl_xor(m1, 16, 32));
  if (lane < 16){ sf[wv][lane]=m0; sf[wv][lane+16]=m1; }
  __syncthreads();
  if (lane < 8){
    float acc = sbs[lane];
    #pragma unroll
    for (int c=0;c<32;++c) acc += sf[wv][c]*sWs[c*8+lane];
    fs[(size_t)(b*512+m)*8 + lane] = lrelu(acc);
  }
}

// ---------------------------------------------------------------------------
// Kernel 4: kNN(K=16) of queries xyz1[::4] (=xyz0[::8], 128) against xyz1
// (=xyz0[::2], 512). Indices are into xyz1 space (0..511).
// ---------------------------------------------------------------------------
__global__ __launch_bounds__(128) void k_knn2(const float* __restrict__ xyz0,
                                              int* __restrict__ idx2)
{
  __shared__ float sx[512], sy[512], sz[512];
  int b = blockIdx.x, t = threadIdx.x;
  for (int j=t;j<512;j+=128){
    const float* p = xyz0 + (size_t)(b*1024 + 2*j)*3;
    sx[j]=p[0]; sy[j]=p[1]; sz[j]=p[2];
  }
  __syncthreads();
  int m = t;                                // 0..127
  float qx=sx[4*m], qy=sy[4*m], qz=sz[4*m];
  float bd[16]; int bi[16];
  #pragma unroll
  for (int k=0;k<16;++k){ bd[k]=3.402823466e38f; bi[k]=0; }
  for (int j=0;j<512;++j){
    float dx=sx[j]-qx, dy=sy[j]-qy, dz=sz[j]-qz;
    float d2 = dx*dx + dy*dy + dz*dz;
    if (d2 < bd[15]){
      bd[15]=d2; bi[15]=j;
      #pragma unroll
      for (int u=14;u>=0;--u){
        if (bd[u+1] < bd[u]){
          float td=bd[u]; bd[u]=bd[u+1]; bd[u+1]=td;
          int ti=bi[u]; bi[u]=bi[u+1]; bi[u+1]=ti;
        }
      }
    }
  }
  int* op = idx2 + (size_t)(b*128+m)*16;
  #pragma unroll
  for (int k=0;k<16;++k) op[k]=bi[k];
}

// ---------------------------------------------------------------------------
// Kernel 5 (WMMA): pointconv2. One wave per (b,m): 11 channels (pad 12),
// 128 outputs -> 3 k-steps x 8 n-tiles of f32 WMMA; lrelu + max over M.
// ---------------------------------------------------------------------------
__global__ __launch_bounds__(128) void k_pc2(
    const float* __restrict__ xyz0, const float* __restrict__ fsin,
    const int* __restrict__ idx2,
    const float* __restrict__ W, const float* __restrict__ bias,
    float* __restrict__ f2)
{
  __shared__ float sW[12*128];   // K padded to 12, zero row
  __shared__ float sbias[128];
  int t = threadIdx.x;
  for (int i=t;i<1536;i+=128){ int c=i>>7, o=i&127; sW[i] = (c<11) ? W[c*128+o] : 0.0f; }
  if (t<128) sbias[t]=bias[t];
  __syncthreads();

  int wv=t>>5, lane=t&31;
  int gw = blockIdx.x*4 + wv;              // 0..B*128-1
  int b = gw>>7, m = gw&127;
  int row = lane&15, kh=(lane>>4)<<1, n=lane&15;

  const float* ctr = xyz0 + (size_t)(b*1024 + 8*m)*3;
  float cx=ctr[0], cy=ctr[1], cz=ctr[2];
  int nb = idx2[(size_t)(b*128+m)*16 + row];          // index into xyz1 (0..511)
  const float* pn = xyz0 + (size_t)(b*1024 + 2*nb)*3;
  const float* fr = fsin + (size_t)(b*512 + nb)*8;
  float gx=pn[0]-cx, gy=pn[1]-cy, gz=pn[2]-cz;

  v8f acc[8];
  #pragma unroll
  for (int tt=0;tt<8;++tt){
    float bv = sbias[tt*16+n];
    #pragma unroll
    for (int i=0;i<8;++i) acc[tt][i]=bv;
  }
  #pragma unroll
  for (int s=0;s<3;++s){
    int c0 = 4*s + kh;
    v2f av; av.x = gchan(c0,   gx,gy,gz, fr, 8);
            av.y = gchan(c0+1, gx,gy,gz, fr, 8);
    #pragma unroll
    for (int tt=0;tt<8;++tt){
      v2f bb; bb.x = sW[c0*128 + tt*16 + n];
              bb.y = sW[(c0+1)*128 + tt*16 + n];
      acc[tt] = wmma4(av, bb, acc[tt]);
    }
  }
  float* out = f2 + (size_t)(b*128+m)*128;
  #pragma unroll
  for (int tt=0;tt<8;++tt){
    float mx = -3.402823466e38f;
    #pragma unroll
    for (int i=0;i<8;++i) mx = fmaxf(mx, lrelu(acc[tt][i]));
    mx = fmaxf(mx, __shfl_xor(mx, 16, 32));
    if (lane < 16) out[tt*16 + lane] = mx;
  }
}

// ---------------------------------------------------------------------------
// Kernel 6: mean over the 128 rows of f2[b] (mean commutes with final matmul)
// ---------------------------------------------------------------------------
__global__ __launch_bounds__(128) void k_avg(const float* __restrict__ f2,
                                             float* __restrict__ favg){
  int b = blockIdx.x, c = threadIdx.x;
  float s = 0.0f;
  for (int m=0;m<128;++m) s += f2[(size_t)(b*128+m)*128 + c];
  favg[b*128+c] = s*(1.0f/128.0f);
}

// ---------------------------------------------------------------------------
// Kernel 7 (WMMA): out = favg(256x128) @ We(128x256) + be.
// One wave per 16x16 output tile, 32 f32 WMMA k-steps.
// ---------------------------------------------------------------------------
__global__ __launch_bounds__(128) void k_final(const float* __restrict__ favg,
                                               const float* __restrict__ We,
                                               const float* __restrict__ be,
                                               float* __restrict__ out){
  int t=threadIdx.x, wv=t>>5, lane=t&31;
  int wave = blockIdx.x*4 + wv;           // 0..255
  int mt = wave>>4, nt = wave&15;
  int n = lane&15, kh=(lane>>4)<<1;
  const float* arow = favg + (size_t)(mt*16 + n)*128;   // A: lane holds row M=lane&15
  int col = nt*16 + n;
  v8f acc; float bv = be[col];
  #pragma unroll
  for (int i=0;i<8;++i) acc[i]=bv;
  #pragma unroll
  for (int s=0;s<32;++s){
    int k = 4*s + kh;
    v2f av; av.x = arow[k]; av.y = arow[k+1];
    v2f bb; bb.x = We[(size_t)k*256 + col]; bb.y = We[(size_t)(k+1)*256 + col];
    acc = wmma4(av, bb, acc);
  }
  int mbase = mt*16 + ((lane>>4)<<3);     // C: VGPR r -> M=r (lanes 0-15) / M=8+r
  #pragma unroll
  for (int r=0;r<8;++r) out[(size_t)(mbase+r)*256 + col] = acc[r];
}

// ---------------------------------------------------------------------------
extern "C" void kernel_launch(void* const* d_in, const int* in_sizes, int n_in,
                              void* d_out, int out_size, void* d_ws, size_t ws_size,
                              hipStream_t stream)
{
  const float* pc   = (const float*)d_in[0];
  const float* W0   = (const float*)d_in[1];
  const float* b0   = (const float*)d_in[2];
  const float* g0   = (const float*)d_in[3];
  const float* be0  = (const float*)d_in[4];
  const float* W1   = (const float*)d_in[5];
  const float* b1   = (const float*)d_in[6];
  const float* g1   = (const float*)d_in[7];
  const float* be1  = (const float*)d_in[8];
  const float* pcW  = (const float*)d_in[9];
  const float* pcb  = (const float*)d_in[10];
  const float* Ws   = (const float*)d_in[11];
  const float* bs   = (const float*)d_in[12];
  const float* Wc   = (const float*)d_in[13];
  const float* bc   = (const float*)d_in[14];
  const float* We   = (const float*)d_in[15];
  const float* beF  = (const float*)d_in[16];
  float* outp = (float*)d_out;
  (void)in_sizes; (void)n_in; (void)out_size; (void)ws_size;

  char* w = (char*)d_ws;
  size_t off = 0;
  auto take = [&](size_t bytes)->void*{
    void* p = w + off;
    off = (off + bytes + 255) & ~(size_t)255;
    return p;
  };
  float* xyz0 = (float*)take((size_t)256*1024*3*sizeof(float));   //  3 MB
  float* feats= (float*)take((size_t)256*1024*16*sizeof(float));  // 16 MB
  int*   idx1 = (int*)  take((size_t)256*512*16*sizeof(int));     //  8 MB
  float* fsb  = (float*)take((size_t)256*512*8*sizeof(float));    //  4 MB
  int*   idx2 = (int*)  take((size_t)256*128*16*sizeof(int));     //  2 MB
  float* f2   = (float*)take((size_t)256*128*128*sizeof(float));  // 16 MB
  float* favg = (float*)take((size_t)256*128*sizeof(float));      // 128 KB

  k_pointwise<<<1024, 256, 0, stream>>>(pc, W0,b0,g0,be0, W1,b1,g1,be1, xyz0, feats);
  k_knn1     <<<256,  256, 0, stream>>>(xyz0, idx1);
  k_pc1      <<<32768,128, 0, stream>>>(xyz0, feats, idx1, pcW, pcb, Ws, bs, fsb);
  k_knn2     <<<256,  128, 0, stream>>>(xyz0, idx2);
  k_pc2      <<<8192, 128, 0, stream>>>(xyz0, fsb, idx2, Wc, bc, f2);
  k_avg      <<<256,  128, 0, stream>>>(f2, favg);
  k_final    <<<64,   128, 0, stream>>>(favg, We, beF, outp);
}